// get_knearest_neighbors_torch_43516608643711
// MI455X (gfx1250) — compile-verified
//
#include <hip/hip_runtime.h>

// KNN (k=16) over 16384 points in 2D, WMMA-accelerated pairwise-distance tiles.
//
// d2[m][j] = sq_m + sq_j - 2*x_m*x_j - 2*y_m*y_j via one V_WMMA_F32_16X16X4_F32
// per 16x16 tile:
//   A[m][0..3] = [-2x_m, -2y_m, sq_m, 1]
//   B[0..3][j] = [  x_j,   y_j,   1 , sq_j]
// f32 WMMA keeps the reference's selection numerics.
//
// Parallelism: rows-per-wave is pinned at 32 by wave32, which alone yields only
// 512 waves (starves an MI455X with 512-1024 SIMD32s). So the column dimension
// is split into SEG=8 stripes handled by independent waves (4096 waves total);
// per-stripe top-17 lists are merged by a second, cheap kernel. Any global
// top-17 element is in its stripe's top-17, so this is exact.
//
// All kernels use one wave32 per workgroup: the LDS staging tile is
// wave-private and __syncthreads() lowers to a free wave-level barrier.

#define NPTS          16384
#define KNB           16
#define KP1           17          // self + 16 neighbors
#define TILE          16
#define ROWS_PER_WAVE 32          // two 16-row WMMA tiles per wave
#define PAD           20          // LDS row stride (floats): 16B-aligned, conflict-free
#define SEG           8           // column stripes per row
#define SEG_COLS      (NPTS / SEG)          // 2048
#define TILES_PER_SEG (SEG_COLS / TILE)     // 128

typedef float v2f __attribute__((ext_vector_type(2)));
typedef float v8f __attribute__((ext_vector_type(8)));

// ---------------------------------------------------------------------------
// Phase 1: per-stripe top-17 (unsorted) -> workspace, transposed for coalescing
// ---------------------------------------------------------------------------
__global__ __launch_bounds__(32)
void knn_wmma_partial(const float* __restrict__ p,
                      float* __restrict__ wsD, int* __restrict__ wsI) {
    __shared__ __align__(16) float tile[ROWS_PER_WAVE][PAD];

    const int lane    = threadIdx.x & 31;
    const int rowBlk  = blockIdx.x / SEG;
    const int seg     = blockIdx.x % SEG;
    const int rowBase = rowBlk * ROWS_PER_WAVE;
    const int colBase = seg * SEG_COLS;

    // A fragments (loop-invariant). 32-bit A 16x4 layout (ISA 7.12.2):
    // lanes 0-15 hold (K=0,K=1) for M=lane; lanes 16-31 hold (K=2,K=3).
    const int mA0 = rowBase + (lane & 15);
    const int mA1 = mA0 + 16;
    const float ax0 = p[mA0 * 3 + 0], ay0 = p[mA0 * 3 + 1];
    const float ax1 = p[mA1 * 3 + 0], ay1 = p[mA1 * 3 + 1];
    v2f a0, a1;
    if (lane < 16) {                      // K=0,1 rows:  -2x, -2y
        a0.x = -2.0f * ax0; a0.y = -2.0f * ay0;
        a1.x = -2.0f * ax1; a1.y = -2.0f * ay1;
    } else {                              // K=2,3 rows:  sq, 1
        a0.x = ax0 * ax0 + ay0 * ay0; a0.y = 1.0f;
        a1.x = ax1 * ax1 + ay1 * ay1; a1.y = 1.0f;
    }

    float bd[KP1]; int bi[KP1];
#pragma unroll
    for (int s = 0; s < KP1; ++s) { bd[s] = __builtin_inff(); bi[s] = 0; }
    float worstv    = __builtin_inff();
    int   worstSlot = 0;

    // Indices ascend within the stripe, so a candidate can never win a
    // lexicographic tie against a buffered entry: `d2 < worstv` is exact.
    auto insert = [&](float d2, int idx) {
#pragma unroll
        for (int s = 0; s < KP1; ++s)
            if (s == worstSlot) { bd[s] = d2; bi[s] = idx; }
        float wv = bd[0]; int wi = bi[0]; int ws = 0;
#pragma unroll
        for (int s = 1; s < KP1; ++s)
            if (bd[s] > wv || (bd[s] == wv && bi[s] > wi)) {
                wv = bd[s]; wi = bi[s]; ws = s;
            }
        worstv = wv; worstSlot = ws;
    };

    for (int ct = 0; ct < TILES_PER_SEG; ++ct) {
        const int C = colBase + ct * TILE;

        // B fragment: 32-bit B 4x16, rows striped across lanes within a VGPR:
        // lanes 0-15 -> (K=0,K=1) = (x_j, y_j); lanes 16-31 -> (K=2,K=3) = (1, sq_j).
        const int   j  = C + (lane & 15);
        const float bx = p[j * 3 + 0], by = p[j * 3 + 1];
        v2f b;
        if (lane < 16) { b.x = bx;   b.y = by; }
        else           { b.x = 1.0f; b.y = bx * bx + by * by; }

        // Prefetch next tile's coordinates (global_prefetch_b8).
        {
            int jn = j + TILE;
            if (jn < NPTS) __builtin_prefetch(p + jn * 3, 0, 3);
        }

        v8f c = {};
        v8f d0 = __builtin_amdgcn_wmma_f32_16x16x4_f32(
            false, a0, false, b, (short)0, c, false, false);
        v8f d1 = __builtin_amdgcn_wmma_f32_16x16x4_f32(
            false, a1, false, b, (short)0, c, false, false);

        // Scatter D into row-major LDS: lanes 0-15: VGPR r -> (M=r, N=lane);
        // lanes 16-31: VGPR r -> (M=r+8, N=lane-16).
        const int col  = lane & 15;
        const int rOff = (lane < 16) ? 0 : 8;
#pragma unroll
        for (int r = 0; r < 8; ++r) {
            tile[rOff + r][col]      = d0[r];
            tile[16 + rOff + r][col] = d1[r];
        }
        __syncthreads();   // single-wave WG: wave-level ordering only

        // Lane L filters the 16 candidates of row rowBase+L.
#pragma unroll
        for (int g = 0; g < 4; ++g) {
            const float4 q = ((const float4*)tile[lane])[g];   // ds_load_b128
            if (q.x < worstv || q.y < worstv || q.z < worstv || q.w < worstv) {
                const int base = C + 4 * g;
                const float qv[4] = {q.x, q.y, q.z, q.w};
#pragma unroll
                for (int t = 0; t < 4; ++t)
                    if (qv[t] < worstv) insert(qv[t], base + t);
            }
        }
        __syncthreads();
    }

    // Dump unsorted partial list, transposed so lanes store consecutively.
    const int myRow = rowBase + lane;
#pragma unroll
    for (int k = 0; k < KP1; ++k) {
        wsD[(seg * KP1 + k) * NPTS + myRow] = bd[k];
        wsI[(seg * KP1 + k) * NPTS + myRow] = bi[k];
    }
}

// ---------------------------------------------------------------------------
// Phase 2: merge SEG partial lists per row, sort, drop self, exact distances
// ---------------------------------------------------------------------------
__global__ __launch_bounds__(32)
void knn_merge(const float* __restrict__ p,
               const float* __restrict__ wsD, const int* __restrict__ wsI,
               float* __restrict__ out) {
    const int lane  = threadIdx.x & 31;
    const int myRow = blockIdx.x * ROWS_PER_WAVE + lane;
    const float myX = p[myRow * 3 + 0];
    const float myY = p[myRow * 3 + 1];

    float bd[KP1]; int bi[KP1];
#pragma unroll
    for (int s = 0; s < KP1; ++s) { bd[s] = __builtin_inff(); bi[s] = 0x7FFFFFFF; }
    float worstv    = __builtin_inff();
    int   worsti    = 0x7FFFFFFF;
    int   worstSlot = 0;

    // Candidates arrive in arbitrary index order across stripes -> full
    // lexicographic (d2, idx) compare to preserve top_k tie-breaking.
    for (int c = 0; c < SEG * KP1; ++c) {        // coalesced b32 loads
        const float d2  = wsD[c * NPTS + myRow];
        const int   idx = wsI[c * NPTS + myRow];
        if (d2 < worstv || (d2 == worstv && idx < worsti)) {
#pragma unroll
            for (int s = 0; s < KP1; ++s)
                if (s == worstSlot) { bd[s] = d2; bi[s] = idx; }
            float wv = bd[0]; int wi = bi[0]; int ws = 0;
#pragma unroll
            for (int s = 1; s < KP1; ++s)
                if (bd[s] > wv || (bd[s] == wv && bi[s] > wi)) {
                    wv = bd[s]; wi = bi[s]; ws = s;
                }
            worstv = wv; worsti = wi; worstSlot = ws;
        }
    }

    // Ordered extraction: rank 0 is self (dropped), ranks 1..16 out.
#pragma unroll
    for (int o = 0; o < KP1; ++o) {
        float mv = bd[0]; int mi = bi[0]; int ms = 0;
#pragma unroll
        for (int s = 1; s < KP1; ++s)
            if (bd[s] < mv || (bd[s] == mv && bi[s] < mi)) {
                mv = bd[s]; mi = bi[s]; ms = s;
            }
#pragma unroll
        for (int s = 0; s < KP1; ++s)
            if (s == ms) bd[s] = __builtin_inff();
        if (o >= 1) {
            const float nx = p[mi * 3 + 0], ny = p[mi * 3 + 1];
            const float dx = myX - nx, dy = myY - ny;
            out[myRow * KNB + (o - 1)] = sqrtf(dx * dx + dy * dy);
        }
    }
}

// ---------------------------------------------------------------------------
// Fallback: single-pass fused kernel (512 waves) if workspace is too small
// ---------------------------------------------------------------------------
__global__ __launch_bounds__(32)
void knn_wmma_fused(const float* __restrict__ p, float* __restrict__ out) {
    __shared__ __align__(16) float tile[ROWS_PER_WAVE][PAD];

    const int lane    = threadIdx.x & 31;
    const int rowBase = blockIdx.x * ROWS_PER_WAVE;

    const int mA0 = rowBase + (lane & 15);
    const int mA1 = mA0 + 16;
    const float ax0 = p[mA0 * 3 + 0], ay0 = p[mA0 * 3 + 1];
    const float ax1 = p[mA1 * 3 + 0], ay1 = p[mA1 * 3 + 1];
    v2f a0, a1;
    if (lane < 16) {
        a0.x = -2.0f * ax0; a0.y = -2.0f * ay0;
        a1.x = -2.0f * ax1; a1.y = -2.0f * ay1;
    } else {
        a0.x = ax0 * ax0 + ay0 * ay0; a0.y = 1.0f;
        a1.x = ax1 * ax1 + ay1 * ay1; a1.y = 1.0f;
    }

    const int   myRow = rowBase + lane;
    const float myX = p[myRow * 3 + 0];
    const float myY = p[myRow * 3 + 1];

    float bd[KP1]; int bi[KP1];
#pragma unroll
    for (int s = 0; s < KP1; ++s) { bd[s] = __builtin_inff(); bi[s] = 0; }
    float worstv    = __builtin_inff();
    int   worstSlot = 0;

    auto insert = [&](float d2, int idx) {
#pragma unroll
        for (int s = 0; s < KP1; ++s)
            if (s == worstSlot) { bd[s] = d2; bi[s] = idx; }
        float wv = bd[0]; int wi = bi[0]; int ws = 0;
#pragma unroll
        for (int s = 1; s < KP1; ++s)
            if (bd[s] > wv || (bd[s] == wv && bi[s] > wi)) {
                wv = bd[s]; wi = bi[s]; ws = s;
            }
        worstv = wv; worstSlot = ws;
    };

    for (int ct = 0; ct < NPTS / TILE; ++ct) {
        const int C = ct * TILE;
        const int   j  = C + (lane & 15);
        const float bx = p[j * 3 + 0], by = p[j * 3 + 1];
        v2f b;
        if (lane < 16) { b.x = bx;   b.y = by; }
        else           { b.x = 1.0f; b.y = bx * bx + by * by; }
        {
            int jn = j + TILE;
            if (jn < NPTS) __builtin_prefetch(p + jn * 3, 0, 3);
        }
        v8f c = {};
        v8f d0 = __builtin_amdgcn_wmma_f32_16x16x4_f32(
            false, a0, false, b, (short)0, c, false, false);
        v8f d1 = __builtin_amdgcn_wmma_f32_16x16x4_f32(
            false, a1, false, b, (short)0, c, false, false);
        const int col  = lane & 15;
        const int rOff = (lane < 16) ? 0 : 8;
#pragma unroll
        for (int r = 0; r < 8; ++r) {
            tile[rOff + r][col]      = d0[r];
            tile[16 + rOff + r][col] = d1[r];
        }
        __syncthreads();
#pragma unroll
        for (int g = 0; g < 4; ++g) {
            const float4 q = ((const float4*)tile[lane])[g];
            if (q.x < worstv || q.y < worstv || q.z < worstv || q.w < worstv) {
                const int base = C + 4 * g;
                const float qv[4] = {q.x, q.y, q.z, q.w};
#pragma unroll
                for (int t = 0; t < 4; ++t)
                    if (qv[t] < worstv) insert(qv[t], base + t);
            }
        }
        __syncthreads();
    }

#pragma unroll
    for (int o = 0; o < KP1; ++o) {
        float mv = bd[0]; int mi = bi[0]; int ms = 0;
#pragma unroll
        for (int s = 1; s < KP1; ++s)
            if (bd[s] < mv || (bd[s] == mv && bi[s] < mi)) {
                mv = bd[s]; mi = bi[s]; ms = s;
            }
#pragma unroll
        for (int s = 0; s < KP1; ++s)
            if (s == ms) bd[s] = __builtin_inff();
        if (o >= 1) {
            const float nx = p[mi * 3 + 0], ny = p[mi * 3 + 1];
            const float dx = myX - nx, dy = myY - ny;
            out[myRow * KNB + (o - 1)] = sqrtf(dx * dx + dy * dy);
        }
    }
}

extern "C" void kernel_launch(void* const* d_in, const int* in_sizes, int n_in,
                              void* d_out, int out_size, void* d_ws, size_t ws_size,
                              hipStream_t stream) {
    (void)in_sizes; (void)n_in; (void)out_size;
    const float* p   = (const float*)d_in[0];
    float*       out = (float*)d_out;

    const size_t listElems = (size_t)SEG * KP1 * NPTS;     // 2,228,224
    const size_t need      = listElems * (sizeof(float) + sizeof(int)); // ~17.8 MB

    if (d_ws != nullptr && ws_size >= need) {
        float* wsD = (float*)d_ws;
        int*   wsI = (int*)(wsD + listElems);
        knn_wmma_partial<<<dim3((NPTS / ROWS_PER_WAVE) * SEG), dim3(32), 0, stream>>>(
            p, wsD, wsI);
        knn_merge<<<dim3(NPTS / ROWS_PER_WAVE), dim3(32), 0, stream>>>(
            p, wsD, wsI, out);
    } else {
        knn_wmma_fused<<<dim3(NPTS / ROWS_PER_WAVE), dim3(32), 0, stream>>>(p, out);
    }
}